// MpcRetrievalLoss_15522011807821
// MI455X (gfx1250) — compile-verified
//
#include <hip/hip_runtime.h>
#include <hip/hip_bf16.h>
#include <math.h>

typedef __attribute__((ext_vector_type(2))) float v2f;
typedef __attribute__((ext_vector_type(8))) float v8f;

#define BDIM 2048
#define DDIM 512
#define KTOT 1024
#define BK   16
#define NTILE (KTOT / BK)   // 64 k-tiles
#define LDSS 20             // padded stride: 80B = 5x16B aligned, gcd(20,64)=4 -> conflict-free

// ---------------- prep query rows ----------------
__global__ __launch_bounds__(256) void prep_q_kernel(
    const float* __restrict__ qmean, const float* __restrict__ qls,
    const float* __restrict__ qz,
    float* __restrict__ Acat, float* __restrict__ colc, float* __restrict__ rq)
{
    __shared__ float red[256];
    const int q = blockIdx.x;
    const int tid = threadIdx.x;
    const float m0 = qmean[q * DDIM + tid];
    const float m1 = qmean[q * DDIM + tid + 256];
    const float l0 = qls[q * DDIM + tid];
    const float l1 = qls[q * DDIM + tid + 256];

    red[tid] = m0 * m0 + m1 * m1;
    __syncthreads();
    for (int s = 128; s > 0; s >>= 1) { if (tid < s) red[tid] += red[tid + s]; __syncthreads(); }
    const float inv = 1.0f / fmaxf(sqrtf(red[0]), 1e-12f);
    __syncthreads();

    red[tid] = l0 + l1;
    __syncthreads();
    for (int s = 128; s > 0; s >>= 1) { if (tid < s) red[tid] += red[tid + s]; __syncthreads(); }
    const float sumls = red[0];
    __syncthreads();

    const float w0 = expf(-l0), w1 = expf(-l1);
    const float n0 = m0 * inv, n1 = m1 * inv;
    Acat[(size_t)q * KTOT + tid]             = w0;
    Acat[(size_t)q * KTOT + tid + 256]       = w1;
    Acat[(size_t)q * KTOT + 512 + tid]       = -2.0f * n0 * w0;
    Acat[(size_t)q * KTOT + 512 + tid + 256] = -2.0f * n1 * w1;

    red[tid] = n0 * n0 * w0 + n1 * n1 * w1;
    __syncthreads();
    for (int s = 128; s > 0; s >>= 1) { if (tid < s) red[tid] += red[tid + s]; __syncthreads(); }
    if (tid == 0) {
        rq[q] = -0.5f * red[0];
        const float norm_const = -256.0f * logf(6.28318530717958647692f) - 0.5f * sumls;
        colc[q] = qz[q] + norm_const;   // torch broadcast quirk: indexed by TARGET axis
    }
}

// ---------------- prep target rows (fold the 7 samples) ----------------
__global__ __launch_bounds__(256) void prep_t_kernel(
    const float* __restrict__ tmean, const float* __restrict__ tls,
    const float* __restrict__ eps, float* __restrict__ Ycat)
{
    __shared__ float red[256];
    const int t = blockIdx.x;
    const int tid = threadIdx.x;
    const float m0 = tmean[t * DDIM + tid];
    const float m1 = tmean[t * DDIM + tid + 256];

    red[tid] = m0 * m0 + m1 * m1;
    __syncthreads();
    for (int s = 128; s > 0; s >>= 1) { if (tid < s) red[tid] += red[tid + s]; __syncthreads(); }
    const float inv = 1.0f / fmaxf(sqrtf(red[0]), 1e-12f);
    __syncthreads();

    #pragma unroll
    for (int h = 0; h < 2; ++h) {
        const int d = tid + h * 256;
        const float tm = tmean[t * DDIM + d] * inv;
        const float sg = expf(tls[t * DDIM + d]);
        float se = 0.0f, se2 = 0.0f;
        #pragma unroll
        for (int s = 0; s < 7; ++s) {
            const float e = eps[(size_t)t * 7 * DDIM + s * DDIM + d];
            se += e; se2 += e * e;
        }
        const float y2 = 7.0f * tm + sg * se;
        const float y1 = 7.0f * tm * tm + 2.0f * tm * sg * se + sg * sg * se2;
        Ycat[(size_t)t * KTOT + d]       = y1;
        Ycat[(size_t)t * KTOT + 512 + d] = y2;
    }
}

// ---------------- fp32 WMMA GEMM, LDS ping-pong double buffering ----------------
// S[q,t] = colc[t] + rq[q] - (1/14) * sum_k Acat[q,k]*Ycat[t,k]
__global__ __launch_bounds__(256) void gemm_kernel(
    const float* __restrict__ Acat, const float* __restrict__ Ycat,
    const float* __restrict__ colc, const float* __restrict__ rq,
    float* __restrict__ S)
{
    __shared__ float lA[2][128 * LDSS];
    __shared__ float lY[2][128 * LDSS];

    const int tid  = threadIdx.x;
    const int lane = tid & 31;
    const int wave = tid >> 5;
    const int wm = wave & 1;        // 2 wave rows   -> 64 rows each
    const int wn = wave >> 1;       // 4 wave cols   -> 32 cols each
    const int lh = lane >> 4;       // half-wave select (K pair / M+8)
    const int ll = lane & 15;
    const int rowBase = blockIdx.y * 128;
    const int colBase = blockIdx.x * 128;

    // cooperative load slots: 128x16 tile = 512 float4 slots / 256 threads = 2 each
    int srow[2], scol[2];
    #pragma unroll
    for (int i = 0; i < 2; ++i) {
        const int f = tid + i * 256;
        srow[i] = f >> 2;            // 0..127
        scol[i] = (f & 3) << 2;      // 0,4,8,12
    }

    v8f acc[4][2] = {};
    float4 ra[2], ry[2];

    // prologue: tile 0 -> buf 0
    #pragma unroll
    for (int i = 0; i < 2; ++i) {
        ra[i] = *(const float4*)&Acat[(size_t)(rowBase + srow[i]) * KTOT + scol[i]];
        ry[i] = *(const float4*)&Ycat[(size_t)(colBase + srow[i]) * KTOT + scol[i]];
    }
    #pragma unroll
    for (int i = 0; i < 2; ++i) {
        *(float4*)&lA[0][srow[i] * LDSS + scol[i]] = ra[i];
        *(float4*)&lY[0][srow[i] * LDSS + scol[i]] = ry[i];
    }
    __syncthreads();

    for (int it = 0; it < NTILE; ++it) {
        const int cur = it & 1;

        // issue global loads for next tile first; their wait lands at the
        // ds_store AFTER the 32 WMMAs, hiding L2 latency behind compute
        const bool more = (it + 1) < NTILE;
        if (more) {
            const int kn = (it + 1) * BK;
            #pragma unroll
            for (int i = 0; i < 2; ++i) {
                ra[i] = *(const float4*)&Acat[(size_t)(rowBase + srow[i]) * KTOT + kn + scol[i]];
                ry[i] = *(const float4*)&Ycat[(size_t)(colBase + srow[i]) * KTOT + kn + scol[i]];
            }
        }

        // compute current tile out of buf[cur]
        const float* __restrict__ cA = lA[cur];
        const float* __restrict__ cY = lY[cur];
        #pragma unroll
        for (int kk = 0; kk < BK; kk += 4) {
            const int ko = kk + (lh << 1);    // lanes 0-15: K=kk,kk+1 ; 16-31: kk+2,kk+3
            v2f af[4], bf[2];
            #pragma unroll
            for (int sm = 0; sm < 4; ++sm)
                af[sm] = *(const v2f*)&cA[(wm * 64 + sm * 16 + ll) * LDSS + ko];
            #pragma unroll
            for (int sn = 0; sn < 2; ++sn)
                bf[sn] = *(const v2f*)&cY[(wn * 32 + sn * 16 + ll) * LDSS + ko];
            #pragma unroll
            for (int sm = 0; sm < 4; ++sm)
                #pragma unroll
                for (int sn = 0; sn < 2; ++sn)
                    acc[sm][sn] = __builtin_amdgcn_wmma_f32_16x16x4_f32(
                        false, af[sm], false, bf[sn],
                        (short)0, acc[sm][sn], false, false);
        }

        // commit next tile to the other buffer; WAR on buf[1-cur] is protected
        // by the previous iteration's barrier (it was last read one iter ago)
        if (more) {
            #pragma unroll
            for (int i = 0; i < 2; ++i) {
                *(float4*)&lA[1 - cur][srow[i] * LDSS + scol[i]] = ra[i];
                *(float4*)&lY[1 - cur][srow[i] * LDSS + scol[i]] = ry[i];
            }
        }
        __syncthreads();   // single barrier per k-tile
    }

    // epilogue: C layout — VGPR r holds M=r (lanes 0-15) / M=r+8 (lanes 16-31), N=lane&15
    #pragma unroll
    for (int sm = 0; sm < 4; ++sm) {
        #pragma unroll
        for (int sn = 0; sn < 2; ++sn) {
            const int col = colBase + wn * 32 + sn * 16 + ll;
            const float cc = colc[col];
            #pragma unroll
            for (int r = 0; r < 8; ++r) {
                const int row = rowBase + wm * 64 + sm * 16 + r + (lh << 3);
                S[(size_t)row * BDIM + col] = cc + rq[row] - acc[sm][sn][r] * (1.0f / 14.0f);
            }
        }
    }
}

// ---------------- column max (over q), two-phase, fixed order ----------------
__global__ __launch_bounds__(256) void colmax_part_kernel(
    const float* __restrict__ S, float* __restrict__ cpart)
{
    const int col = blockIdx.x * 256 + threadIdx.x;
    const int r0  = blockIdx.y * 128;
    float m = -3.402823466e38f;
    for (int r = 0; r < 128; ++r)
        m = fmaxf(m, S[(size_t)(r0 + r) * BDIM + col]);
    cpart[(size_t)blockIdx.y * BDIM + col] = m;
}

__global__ __launch_bounds__(256) void colmax_red_kernel(
    const float* __restrict__ cpart, float* __restrict__ colmax)
{
    const int col = blockIdx.x * 256 + threadIdx.x;
    float m = -3.402823466e38f;
    for (int i = 0; i < 16; ++i)
        m = fmaxf(m, cpart[(size_t)i * BDIM + col]);
    colmax[col] = m;
}

// ---------------- per-row stable log-softmax, keep diagonal ----------------
__global__ __launch_bounds__(256) void row_lse_kernel(
    const float* __restrict__ S, const float* __restrict__ colmax,
    float* __restrict__ rowvals)
{
    __shared__ float red[256];
    const int q = blockIdx.x;
    const int tid = threadIdx.x;
    const float* row = S + (size_t)q * BDIM;

    float m = -3.402823466e38f;
    for (int t = tid; t < BDIM; t += 256)
        m = fmaxf(m, row[t] - colmax[t]);
    red[tid] = m;
    __syncthreads();
    for (int s = 128; s > 0; s >>= 1) { if (tid < s) red[tid] = fmaxf(red[tid], red[tid + s]); __syncthreads(); }
    m = red[0];
    __syncthreads();

    float sum = 0.0f;
    for (int t = tid; t < BDIM; t += 256)
        sum += expf(row[t] - colmax[t] - m);
    red[tid] = sum;
    __syncthreads();
    for (int s = 128; s > 0; s >>= 1) { if (tid < s) red[tid] += red[tid + s]; __syncthreads(); }

    if (tid == 0)
        rowvals[q] = (row[q] - colmax[q]) - (m + logf(red[0]));  // logp[q,q]
}

__global__ __launch_bounds__(256) void final_loss_kernel(
    const float* __restrict__ rowvals, float* __restrict__ out)
{
    __shared__ float red[256];
    const int tid = threadIdx.x;
    float s = 0.0f;
    for (int i = tid; i < BDIM; i += 256) s += rowvals[i];
    red[tid] = s;
    __syncthreads();
    for (int k = 128; k > 0; k >>= 1) { if (tid < k) red[tid] += red[tid + k]; __syncthreads(); }
    if (tid == 0) out[0] = -red[0] / (float)BDIM;
}

extern "C" void kernel_launch(void* const* d_in, const int* in_sizes, int n_in,
                              void* d_out, int out_size, void* d_ws, size_t ws_size,
                              hipStream_t stream) {
    const float* query_mean     = (const float*)d_in[0];   // [B,D]
    const float* query_logsigma = (const float*)d_in[1];   // [B,D]
    const float* query_z        = (const float*)d_in[2];   // [B]
    const float* target_mean    = (const float*)d_in[3];   // [B,D]
    const float* target_logsigma= (const float*)d_in[4];   // [B,D]
    // d_in[5] = target_z: dead in the reference (broadcast quirk uses query_z on t-axis)
    const float* eps            = (const float*)d_in[6];   // [B,7,D]
    float* out = (float*)d_out;

    float* ws    = (float*)d_ws;
    float* Acat  = ws;                                   // B*1024
    float* Ycat  = Acat + (size_t)BDIM * KTOT;           // B*1024
    float* S     = Ycat + (size_t)BDIM * KTOT;           // B*B
    float* colc  = S + (size_t)BDIM * BDIM;              // B
    float* rq    = colc + BDIM;                          // B
    float* colmax = rq + BDIM;                           // B
    float* rowvals = colmax + BDIM;                      // B
    float* cpart = rowvals + BDIM;                       // 16*B

    prep_q_kernel<<<BDIM, 256, 0, stream>>>(query_mean, query_logsigma, query_z,
                                            Acat, colc, rq);
    prep_t_kernel<<<BDIM, 256, 0, stream>>>(target_mean, target_logsigma, eps, Ycat);

    dim3 ggrid(BDIM / 128, BDIM / 128);
    gemm_kernel<<<ggrid, 256, 0, stream>>>(Acat, Ycat, colc, rq, S);

    dim3 cgrid(BDIM / 256, 16);
    colmax_part_kernel<<<cgrid, 256, 0, stream>>>(S, cpart);
    colmax_red_kernel<<<BDIM / 256, 256, 0, stream>>>(cpart, colmax);

    row_lse_kernel<<<BDIM, 256, 0, stream>>>(S, colmax, rowvals);
    final_loss_kernel<<<1, 256, 0, stream>>>(rowvals, out);
}